// HierarchicalConsistencyLoss_84241488544386
// MI455X (gfx1250) — compile-verified
//
#include <hip/hip_runtime.h>
#include <hip/hip_bf16.h>

#define T_BINS 1000

typedef __attribute__((ext_vector_type(2))) float v2f;
typedef __attribute__((ext_vector_type(8))) float v8f;

// ---------------------------------------------------------------------------
// Stage 0: zero the [T][4] accumulator region of the workspace.
// ---------------------------------------------------------------------------
__global__ void hcl_zero_ws(float* __restrict__ ws, int n) {
    int i = blockIdx.x * blockDim.x + threadIdx.x;
    if (i < n) ws[i] = 0.0f;
}

// ---------------------------------------------------------------------------
// Stage 1: streaming segment-sum into per-block LDS histogram (ds_add_f32),
// flushed once per block with global_atomic_add_f32.
// Each thread handles 4 consecutive points -> b128 loads on all streams.
// ---------------------------------------------------------------------------
__global__ __launch_bounds__(256) void hcl_stage1(
    const float* __restrict__ coords,
    const float* __restrict__ oinst,
    const float* __restrict__ otree,
    const int*   __restrict__ labels,
    float* __restrict__ ws,   // [T_BINS][4] : sx, sy, sz, cnt
    int n)
{
    __shared__ float h[T_BINS * 4];
    for (int i = threadIdx.x; i < T_BINS * 4; i += blockDim.x) h[i] = 0.0f;
    __syncthreads();

    const int gid    = blockIdx.x * blockDim.x + threadIdx.x;
    const int stride = gridDim.x * blockDim.x;   // in units of 4-point groups
    const int n4     = n >> 2;

    const float4* __restrict__ cc = (const float4*)coords;
    const float4* __restrict__ ii = (const float4*)oinst;
    const float4* __restrict__ tt = (const float4*)otree;
    const int4*   __restrict__ ll = (const int4*)labels;

    for (int p4 = gid; p4 < n4; p4 += stride) {
        // prefetch next tile (speculative, gfx1250 global_prefetch_b8)
        __builtin_prefetch(&cc[(size_t)(p4 + stride) * 3], 0, 0);
        __builtin_prefetch(&ii[(size_t)(p4 + stride) * 3], 0, 0);
        __builtin_prefetch(&tt[(size_t)(p4 + stride) * 3], 0, 0);
        __builtin_prefetch(&ll[p4 + stride], 0, 0);

        float4 c0 = cc[(size_t)p4 * 3 + 0];
        float4 c1 = cc[(size_t)p4 * 3 + 1];
        float4 c2 = cc[(size_t)p4 * 3 + 2];
        float4 a0 = ii[(size_t)p4 * 3 + 0];
        float4 a1 = ii[(size_t)p4 * 3 + 1];
        float4 a2 = ii[(size_t)p4 * 3 + 2];
        float4 b0 = tt[(size_t)p4 * 3 + 0];
        float4 b1 = tt[(size_t)p4 * 3 + 1];
        float4 b2 = tt[(size_t)p4 * 3 + 2];
        int4   lv = ll[p4];

        float c[12] = {c0.x,c0.y,c0.z,c0.w, c1.x,c1.y,c1.z,c1.w, c2.x,c2.y,c2.z,c2.w};
        float a[12] = {a0.x,a0.y,a0.z,a0.w, a1.x,a1.y,a1.z,a1.w, a2.x,a2.y,a2.z,a2.w};
        float b[12] = {b0.x,b0.y,b0.z,b0.w, b1.x,b1.y,b1.z,b1.w, b2.x,b2.y,b2.z,b2.w};
        int   l[4]  = {lv.x, lv.y, lv.z, lv.w};

        #pragma unroll
        for (int j = 0; j < 4; ++j) {
            int t = l[j];
            if (t > 0 && t < T_BINS) {
                int k = 3 * j;
                // match reference numerics: (c + oi) - (c + ot)
                float dx = (c[k+0] + a[k+0]) - (c[k+0] + b[k+0]);
                float dy = (c[k+1] + a[k+1]) - (c[k+1] + b[k+1]);
                float dz = (c[k+2] + a[k+2]) - (c[k+2] + b[k+2]);
                float* e = &h[t * 4];
                unsafeAtomicAdd(&e[0], dx);
                unsafeAtomicAdd(&e[1], dy);
                unsafeAtomicAdd(&e[2], dz);
                unsafeAtomicAdd(&e[3], 1.0f);
            }
        }
    }

    // scalar tail (n % 4 points) handled by the first few global threads
    int rem = n & 3;
    if (gid < rem) {
        int p = n - rem + gid;
        int t = labels[p];
        if (t > 0 && t < T_BINS) {
            float dx = (coords[3*p+0] + oinst[3*p+0]) - (coords[3*p+0] + otree[3*p+0]);
            float dy = (coords[3*p+1] + oinst[3*p+1]) - (coords[3*p+1] + otree[3*p+1]);
            float dz = (coords[3*p+2] + oinst[3*p+2]) - (coords[3*p+2] + otree[3*p+2]);
            float* e = &h[t * 4];
            unsafeAtomicAdd(&e[0], dx);
            unsafeAtomicAdd(&e[1], dy);
            unsafeAtomicAdd(&e[2], dz);
            unsafeAtomicAdd(&e[3], 1.0f);
        }
    }

    __syncthreads();

    // flush block-local histogram to global accumulator (skip zeros)
    for (int i = threadIdx.x; i < T_BINS * 4; i += blockDim.x) {
        float v = h[i];
        if (v != 0.0f) unsafeAtomicAdd(&ws[i], v);
    }
}

// ---------------------------------------------------------------------------
// Stage 2: single wave32. Per-lane partials over trees, then full 32-lane
// reduction via V_WMMA_F32_16X16X4_F32 (A = lane values, B = ones):
// D[m][n] = sum_k A[m][k] -> row sums cover all 32 lanes; each lane sums its
// 8 D VGPRs (rows 0-7 in lanes<16, rows 8-15 in lanes>=16), one shfl_xor(16)
// completes the sum.
// ---------------------------------------------------------------------------
__device__ inline float hcl_wave_sum_wmma(float v) {
    v2f a; a.x = v;    a.y = 0.0f;   // lane L: A[L%16][K0 or K2] = v
    v2f b; b.x = 1.0f; b.y = 1.0f;   // B = ones (4x16)
    v8f c = {};
    v8f d = __builtin_amdgcn_wmma_f32_16x16x4_f32(
        /*neg_a=*/false, a, /*neg_b=*/false, b,
        /*c_mod=*/(short)0, c, /*reuse_a=*/false, /*reuse_b=*/false);
    float s = d[0] + d[1] + d[2] + d[3] + d[4] + d[5] + d[6] + d[7];
    s += __shfl_xor(s, 16, 32);
    return s;   // replicated in all lanes
}

__global__ __launch_bounds__(32) void hcl_stage2(
    const float* __restrict__ ws,  // [T_BINS][4]
    float* __restrict__ out)
{
    const int lane = threadIdx.x;
    float total  = 0.0f;
    float ntrees = 0.0f;

    const float4* __restrict__ s4 = (const float4*)ws;
    for (int t = lane; t < T_BINS; t += 32) {
        float4 s = s4[t];
        float cnt = s.w;
        if (cnt >= 1.0f) ntrees += 1.0f;
        if (cnt >= 2.0f) {
            float inv = 1.0f / cnt;            // counts >= 2 => max(cnt,1)==cnt
            float mx = s.x * inv, my = s.y * inv, mz = s.z * inv;
            total += mx * mx + my * my + mz * mz;
        }
    }

    float tot_all = hcl_wave_sum_wmma(total);
    float ntr_all = hcl_wave_sum_wmma(ntrees);

    if (lane == 0) {
        out[0] = (ntr_all > 0.0f) ? (tot_all / ntr_all) : 0.0f;
    }
}

// ---------------------------------------------------------------------------
extern "C" void kernel_launch(void* const* d_in, const int* in_sizes, int n_in,
                              void* d_out, int out_size, void* d_ws, size_t ws_size,
                              hipStream_t stream) {
    const float* coords = (const float*)d_in[0];
    const float* oinst  = (const float*)d_in[1];
    const float* otree  = (const float*)d_in[2];
    const int*   labels = (const int*)d_in[3];
    float*       out    = (float*)d_out;
    float*       ws     = (float*)d_ws;

    const int n = in_sizes[3];   // number of points (labels is [N])

    hcl_zero_ws<<<(T_BINS * 4 + 255) / 256, 256, 0, stream>>>(ws, T_BINS * 4);

    const int blocks = 512;      // ~8 grid-stride iters/thread; flush = 2M atomics (~8MB)
    hcl_stage1<<<blocks, 256, 0, stream>>>(coords, oinst, otree, labels, ws, n);

    hcl_stage2<<<1, 32, 0, stream>>>(ws, out);
}